// droneDeliveryModel_37692632990431
// MI455X (gfx1250) — compile-verified
//
#include <hip/hip_runtime.h>

typedef __attribute__((ext_vector_type(2))) float v2f;
typedef __attribute__((ext_vector_type(8))) float v8f;

static constexpr int kN = 100000;   // nodes (divisible by 16)
static constexpr int kE = 1600000;  // edges (divisible by 256)
static constexpr int kC = 64;       // hidden channels
static constexpr int kCO = 8;       // output channels

// ---------------------------------------------------------------- utilities
__global__ void zero_f4_kernel(float4* __restrict__ p) {
  int i = blockIdx.x * blockDim.x + threadIdx.x;
  p[i] = make_float4(0.f, 0.f, 0.f, 0.f);
}

__global__ void zero_i_kernel(int* __restrict__ p, int n) {
  int i = blockIdx.x * blockDim.x + threadIdx.x;
  if (i < n) p[i] = 0;
}

// ------------------------------------------------------- degree computation
__global__ void degree_kernel(const int* __restrict__ dst, int* __restrict__ cnt) {
  int e = blockIdx.x * blockDim.x + threadIdx.x;  // grid sized exactly to kE
  atomicAdd(&cnt[dst[e]], 1);
}

// ----------------------------------------------------- neighbor scatter-add
// 16 threads per edge, float4 per thread; L2-resident feature matrix means
// these hardware f32 atomics run at L2 bandwidth.
__global__ void scatter_kernel(const float* __restrict__ hin,
                               const int* __restrict__ src,
                               const int* __restrict__ dst,
                               float* __restrict__ agg) {
  int t = blockIdx.x * blockDim.x + threadIdx.x;  // grid sized to kE*16
  int e = t >> 4;
  int c4 = t & 15;
  int s = src[e];
  int d = dst[e];
  float4 v = reinterpret_cast<const float4*>(hin + (size_t)s * kC)[c4];
  float* ap = agg + (size_t)d * kC + c4 * 4;
  unsafeAtomicAdd(ap + 0, v.x);
  unsafeAtomicAdd(ap + 1, v.y);
  unsafeAtomicAdd(ap + 2, v.z);
  unsafeAtomicAdd(ap + 3, v.w);
}

// -------------------------------------------------- fused SAGE layer (WMMA)
// hout[n,co] = relu( (agg[n,:]/max(cnt,1)) @ Wl^T + hin[n,:] @ Wr^T + bl )
// Block = 128 threads = 4 waves; wave w owns column tile [16w, 16w+16);
// block b owns node tile [16b, 16b+16). K=64 -> 16 steps of V_WMMA_F32_16X16X4_F32.
__global__ void __launch_bounds__(128)
sage_layer_kernel(const float* __restrict__ hin, const float* __restrict__ agg,
                  const int* __restrict__ cnt,
                  const float* __restrict__ Wl, const float* __restrict__ bl,
                  const float* __restrict__ Wr, float* __restrict__ hout) {
  const int lane = threadIdx.x & 31;
  const int wave = threadIdx.x >> 5;
  const int colBase = wave * 16;
  const int nodeBase = blockIdx.x * 16;
  const int m = lane & 15;       // A row == B column index for this lane
  const int half = lane >> 4;    // 0: K pair {0,1}; 1: K pair {2,3}
  const int kk = half * 2;

  const int node = nodeBase + m;
  const float inv = 1.0f / fmaxf((float)cnt[node], 1.0f);
  const float* __restrict__ arow = agg + (size_t)node * kC;
  const float* __restrict__ hrow = hin + (size_t)node * kC;
  const float* __restrict__ wl = Wl + (size_t)(colBase + m) * kC;  // row-major (cout,cin)
  const float* __restrict__ wr = Wr + (size_t)(colBase + m) * kC;

  v8f accL = {};  // neighbor (mean) path
  v8f accR = {};  // root path
#pragma unroll
  for (int k0 = 0; k0 < kC; k0 += 4) {
    v2f a, b;
    a.x = arow[k0 + kk] * inv;
    a.y = arow[k0 + kk + 1] * inv;
    b.x = wl[k0 + kk];
    b.y = wl[k0 + kk + 1];
    accL = __builtin_amdgcn_wmma_f32_16x16x4_f32(false, a, false, b,
                                                 (short)0, accL, false, false);
    a.x = hrow[k0 + kk];
    a.y = hrow[k0 + kk + 1];
    b.x = wr[k0 + kk];
    b.y = wr[k0 + kk + 1];
    accR = __builtin_amdgcn_wmma_f32_16x16x4_f32(false, a, false, b,
                                                 (short)0, accR, false, false);
  }

  const float bias = bl[colBase + m];
#pragma unroll
  for (int r = 0; r < 8; ++r) {
    const int row = r + half * 8;  // D layout: VGPR r -> row r (lo lanes) / r+8 (hi lanes)
    const float v = accL[r] + accR[r] + bias;
    hout[(size_t)(nodeBase + row) * kC + colBase + m] = fmaxf(v, 0.0f);
  }
}

// -------------------------------------------- output projection (WMMA, 8 cols)
__global__ void __launch_bounds__(32)
out_proj_kernel(const float* __restrict__ hin, const float* __restrict__ Wo,
                const float* __restrict__ bo, float* __restrict__ out) {
  const int lane = threadIdx.x & 31;
  const int nodeBase = blockIdx.x * 16;
  const int m = lane & 15;
  const int half = lane >> 4;
  const int kk = half * 2;

  const int node = nodeBase + m;
  const float* __restrict__ hrow = hin + (size_t)node * kC;
  const float mask = (m < kCO) ? 1.0f : 0.0f;
  const float* __restrict__ wo = Wo + (size_t)(m < kCO ? m : 0) * kC;  // clamped, masked below

  v8f acc = {};
#pragma unroll
  for (int k0 = 0; k0 < kC; k0 += 4) {
    v2f a, b;
    a.x = hrow[k0 + kk];
    a.y = hrow[k0 + kk + 1];
    b.x = wo[k0 + kk] * mask;
    b.y = wo[k0 + kk + 1] * mask;
    acc = __builtin_amdgcn_wmma_f32_16x16x4_f32(false, a, false, b,
                                                (short)0, acc, false, false);
  }

  if (m < kCO) {
    const float bias = bo[m];
#pragma unroll
    for (int r = 0; r < 8; ++r) {
      const int outNode = nodeBase + r + half * 8;
      if (outNode < kN - 1) {  // reference drops the last node
        out[(size_t)outNode * kCO + m] = acc[r] + bias;
      }
    }
  }
}

// --------------------------------------------------------------------- host
extern "C" void kernel_launch(void* const* d_in, const int* in_sizes, int n_in,
                              void* d_out, int out_size, void* d_ws, size_t ws_size,
                              hipStream_t stream) {
  const float* x   = (const float*)d_in[0];
  const int*   ei  = (const int*)d_in[1];
  const int*   src = ei;
  const int*   dst = ei + kE;
  const float* Wl1 = (const float*)d_in[2];
  const float* bl1 = (const float*)d_in[3];
  const float* Wr1 = (const float*)d_in[4];
  const float* Wl2 = (const float*)d_in[5];
  const float* bl2 = (const float*)d_in[6];
  const float* Wr2 = (const float*)d_in[7];
  const float* Wl3 = (const float*)d_in[8];
  const float* bl3 = (const float*)d_in[9];
  const float* Wr3 = (const float*)d_in[10];
  const float* Wo  = (const float*)d_in[11];
  const float* bo  = (const float*)d_in[12];
  float* out = (float*)d_out;

  // workspace layout (256B aligned regions)
  char* ws = (char*)d_ws;
  size_t off = 0;
  int* cnt = (int*)(ws + off);
  off += ((size_t)kN * sizeof(int) + 255) & ~(size_t)255;
  float* agg = (float*)(ws + off);
  off += (size_t)kN * kC * sizeof(float);
  float* hA = (float*)(ws + off);
  off += (size_t)kN * kC * sizeof(float);
  float* hB = (float*)(ws + off);
  (void)ws_size; (void)n_in; (void)in_sizes; (void)out_size;

  const int nodeTiles = kN / 16;       // 6250
  const int zeroAggBlk = kN * 16 / 256;  // 6250 float4-stores
  const int degBlk = kE / 256;         // 6250
  const int scatBlk = kE * 16 / 256;   // 100000

  // degree counts (shared by all layers)
  zero_i_kernel<<<(kN + 255) / 256, 256, 0, stream>>>(cnt, kN);
  degree_kernel<<<degBlk, 256, 0, stream>>>(dst, cnt);

  // layer 1: x -> hA
  zero_f4_kernel<<<zeroAggBlk, 256, 0, stream>>>((float4*)agg);
  scatter_kernel<<<scatBlk, 256, 0, stream>>>(x, src, dst, agg);
  sage_layer_kernel<<<nodeTiles, 128, 0, stream>>>(x, agg, cnt, Wl1, bl1, Wr1, hA);

  // layer 2: hA -> hB
  zero_f4_kernel<<<zeroAggBlk, 256, 0, stream>>>((float4*)agg);
  scatter_kernel<<<scatBlk, 256, 0, stream>>>(hA, src, dst, agg);
  sage_layer_kernel<<<nodeTiles, 128, 0, stream>>>(hA, agg, cnt, Wl2, bl2, Wr2, hB);

  // layer 3: hB -> hA
  zero_f4_kernel<<<zeroAggBlk, 256, 0, stream>>>((float4*)agg);
  scatter_kernel<<<scatBlk, 256, 0, stream>>>(hB, src, dst, agg);
  sage_layer_kernel<<<nodeTiles, 128, 0, stream>>>(hB, agg, cnt, Wl3, bl3, Wr3, hA);

  // output projection (drops last node)
  out_proj_kernel<<<nodeTiles, 32, 0, stream>>>(hA, Wo, bo, out);
}